// DETRLogicLoss_26379689132583
// MI455X (gfx1250) — compile-verified
//
#include <hip/hip_runtime.h>
#include <math.h>

#define BB   16
#define QQ   900
#define CC   80
#define NROW (BB * QQ)          // 14400
#define NTILE (NROW / 16)       // 900 WMMA tiles
#define KCH  4                  // k-chunks for argmax split
#define KLEN (QQ / KCH)         // 225
#define CONF_TH (-1.38629436111989062f)   // ln(0.2/0.8)
#define EPSF 1e-7f

typedef __attribute__((ext_vector_type(2))) float v2f;
typedef __attribute__((ext_vector_type(8))) float v8f;

__device__ __forceinline__ float sigmoidf_(float x) {
    // fast reciprocal: v_rcp_f32 instead of IEEE div_fixup chain
    return __builtin_amdgcn_rcpf(1.0f + __expf(-x));
}
__device__ __forceinline__ float softplusf_(float x) {
    // stable softplus with native v_log_f32/v_exp_f32
    return fmaxf(x, 0.0f) + __logf(1.0f + __expf(-fabsf(x)));
}

// deterministic fixed-order tree reduction over a 256-thread block
__device__ __forceinline__ float block_reduce256(float v, float* red) {
    red[threadIdx.x] = v;
    __syncthreads();
    #pragma unroll
    for (int s = 128; s > 0; s >>= 1) {
        if (threadIdx.x < (unsigned)s) red[threadIdx.x] += red[threadIdx.x + s];
        __syncthreads();
    }
    return red[0];
}

// ---------------------------------------------------------------------------
// conf0[b] = (max_c t_logits[b,0,c] > ln(0.25)); also zero conf_any.
// ---------------------------------------------------------------------------
__global__ void k_conf0(const float* __restrict__ tlog,
                        int* __restrict__ conf0, int* __restrict__ conf_any) {
    int b = threadIdx.x;
    if (b == 0) *conf_any = 0;
    if (b < BB) {
        const float* p = tlog + (size_t)b * QQ * CC;   // row q=0 of batch b
        float m = -1e30f;
        for (int c = 0; c < CC; ++c) m = fmaxf(m, p[c]);
        conf0[b] = (m > CONF_TH) ? 1 : 0;
    }
}

// ---------------------------------------------------------------------------
// conf_any = OR over all (b,q,c) of (t_logit > ln(0.25)); float4 scan.
// ---------------------------------------------------------------------------
__global__ void k_confany(const float4* __restrict__ tlog4, int* __restrict__ conf_any) {
    __shared__ int any;
    if (threadIdx.x == 0) any = 0;
    __syncthreads();
    size_t i = (size_t)blockIdx.x * blockDim.x + threadIdx.x;
    if (i < (size_t)NROW * CC / 4) {
        float4 v = tlog4[i];
        float m = fmaxf(fmaxf(v.x, v.y), fmaxf(v.z, v.w));
        if (m > CONF_TH) any = 1;
    }
    __syncthreads();
    if (threadIdx.x == 0 && any) atomicOr(conf_any, 1);
}

// ---------------------------------------------------------------------------
// Partial argmax over a K-chunk of 225 t-boxes, division-free compare.
// grid = BB(16) * rowchunk(4) * kchunk(4) = 256 blocks of 256 threads.
// Candidates (inter, union, idx) written per (kchunk, row).
// ---------------------------------------------------------------------------
__global__ void k_argmax_part(const float* __restrict__ sbox, const float* __restrict__ tbox,
                              float* __restrict__ cand_i, float* __restrict__ cand_u,
                              int* __restrict__ cand_j) {
    const int blk = blockIdx.x;
    const int b  = blk >> 4;
    const int rc = (blk >> 2) & 3;
    const int kc = blk & 3;

    __shared__ float tx1[KLEN], tx2[KLEN], ty1[KLEN], ty2[KLEN], tar[KLEN];
    for (int k = threadIdx.x; k < KLEN; k += blockDim.x) {
        const float* p = tbox + ((size_t)b * QQ + kc * KLEN + k) * 4;
        float x = p[0], y = p[1], w = p[2], h = p[3];
        tx1[k] = x - 0.5f * w; tx2[k] = x + 0.5f * w;
        ty1[k] = y - 0.5f * h; ty2[k] = y + 0.5f * h;
        tar[k] = w * h;
    }
    __syncthreads();

    const int q = rc * 256 + threadIdx.x;
    if (q >= QQ) return;

    const float* p = sbox + ((size_t)b * QQ + q) * 4;
    float x = p[0], y = p[1], w = p[2], h = p[3];
    const float sx1 = x - 0.5f * w, sx2 = x + 0.5f * w;
    const float sy1 = y - 0.5f * h, sy2 = y + 0.5f * h;
    const float sar = w * h;

    float bi = -1.0f, bu = 1.0f;                 // first candidate always wins
    int   bidx = kc * KLEN;
    for (int k = 0; k < KLEN; ++k) {
        float ix = fmaxf(fminf(sx2, tx2[k]) - fmaxf(sx1, tx1[k]), 0.0f);
        float iy = fmaxf(fminf(sy2, ty2[k]) - fmaxf(sy1, ty1[k]), 0.0f);
        float inter = ix * iy;
        float un = sar + tar[k] - inter + EPSF;  // strictly positive
        bool better = inter * bu > bi * un;      // iou_k > iou_best, first-max kept
        bi   = better ? inter : bi;
        bu   = better ? un    : bu;
        bidx = better ? (kc * KLEN + k) : bidx;
    }
    const int row = b * QQ + q;
    cand_i[kc * NROW + row] = bi;
    cand_u[kc * NROW + row] = bu;
    cand_j[kc * NROW + row] = bidx;
}

// ---------------------------------------------------------------------------
// Merge the 4 per-chunk candidates; emit max_idx and iou_ok (iou > 0.1).
// Strict > with ascending chunk order preserves jnp.argmax first-max rule.
// ---------------------------------------------------------------------------
__global__ void k_argmax_merge(const float* __restrict__ cand_i, const float* __restrict__ cand_u,
                               const int* __restrict__ cand_j,
                               int* __restrict__ max_idx, int* __restrict__ iou_ok) {
    const int t = blockIdx.x * 256 + threadIdx.x;
    if (t >= NROW) return;
    float bi = cand_i[t], bu = cand_u[t];
    int   bj = cand_j[t];
    #pragma unroll
    for (int c = 1; c < KCH; ++c) {
        float ci = cand_i[c * NROW + t];
        float cu = cand_u[c * NROW + t];
        int   cj = cand_j[c * NROW + t];
        bool better = ci * bu > bi * cu;
        bi = better ? ci : bi;
        bu = better ? cu : bu;
        bj = better ? cj : bj;
    }
    max_idx[t] = bj;
    iou_ok[t]  = (bi > 0.1f * bu) ? 1 : 0;
}

// ---------------------------------------------------------------------------
// lcls partials: per 16-row tile, diag(S · T_gatheredᵀ) via f32 WMMA 16x16x4
// (fully unrolled K: 20 v_wmma per tile), fused softplus row-sum.
// Block = 256 threads = 8 waves; one tile per wave.
// t-row gather uses batch-0 rows only (replicates reference's flat gather).
// ---------------------------------------------------------------------------
__global__ void k_lcls(const float* __restrict__ slog, const float* __restrict__ tlog,
                       const int* __restrict__ max_idx, float* __restrict__ part) {
    __shared__ float red[256];
    const int wave = threadIdx.x >> 5;
    const int lane = threadIdx.x & 31;
    const int tile = blockIdx.x * 8 + wave;

    float partial = 0.0f;
    if (tile < NTILE) {
        const int r    = lane & 15;      // matrix row handled by this lane
        const int half = lane >> 4;      // 0: K%4 in {0,1}; 1: K%4 in {2,3}
        const int fr   = tile * 16 + r;  // flat (b*Q+q) row
        const float* srow = slog + (size_t)fr * CC;
        const float* trow = tlog + (size_t)max_idx[fr] * CC;   // batch-0 gather
        __builtin_prefetch(trow, 0, 0);

        float sp = 0.0f;
        v8f c = {};
        #pragma unroll
        for (int k0 = 0; k0 < CC; k0 += 4) {
            const int c0 = k0 + half * 2;
            v2f a;  a.x = srow[c0];  a.y = srow[c0 + 1];
            v2f bm; bm.x = sigmoidf_(trow[c0]); bm.y = sigmoidf_(trow[c0 + 1]);
            sp += softplusf_(a.x) + softplusf_(a.y);
            c = __builtin_amdgcn_wmma_f32_16x16x4_f32(
                    false, a, false, bm, (short)0, c, false, false);
        }
        // diagonal of 16x16 f32 D: M==N at lanes 0..7 (vgpr=lane) and
        // lanes 24..31 (vgpr=lane-24).
        union { v8f v; float f[8]; } u; u.v = c;
        float d = 0.0f;
        if (lane < 8)        d = u.f[lane];
        else if (lane >= 24) d = u.f[lane - 24];
        partial = (sp - d) * (1.0f / (float)CC);
    }
    float s = block_reduce256(partial, red);
    if (threadIdx.x == 0) part[blockIdx.x] = s;
}

// ---------------------------------------------------------------------------
// Box-loss partials: per (b,q) gather j=max_idx, flag = iou_ok[b,j] & conf0[b];
// L1 and (1 - GIoU) between s_box[b,j] and t_box[b,0].
// ---------------------------------------------------------------------------
__global__ void k_box(const float* __restrict__ sbox, const float* __restrict__ tbox,
                      const int* __restrict__ max_idx, const int* __restrict__ iou_ok,
                      const int* __restrict__ conf0,
                      float* __restrict__ l1part, float* __restrict__ gpart) {
    __shared__ float red[256];
    const int t = blockIdx.x * 256 + threadIdx.x;
    float l1v = 0.0f, gv = 0.0f;
    if (t < NROW) {
        const int b = t / QQ;
        const int j = max_idx[t];
        if (iou_ok[b * QQ + j] && conf0[b]) {
            const float* ps = sbox + ((size_t)b * QQ + j) * 4;
            const float* pt = tbox + (size_t)b * QQ * 4;      // t row 0
            float x1 = ps[0], y1 = ps[1], w1 = ps[2], h1 = ps[3];
            float x2 = pt[0], y2 = pt[1], w2 = pt[2], h2 = pt[3];
            l1v = fabsf(x1 - x2) + fabsf(y1 - y2) + fabsf(w1 - w2) + fabsf(h1 - h2);
            float a1x1 = x1 - 0.5f * w1, a1x2 = x1 + 0.5f * w1;
            float a1y1 = y1 - 0.5f * h1, a1y2 = y1 + 0.5f * h1;
            float a2x1 = x2 - 0.5f * w2, a2x2 = x2 + 0.5f * w2;
            float a2y1 = y2 - 0.5f * h2, a2y2 = y2 + 0.5f * h2;
            float iw = fmaxf(fminf(a1x2, a2x2) - fmaxf(a1x1, a2x1), 0.0f);
            float ih = fmaxf(fminf(a1y2, a2y2) - fmaxf(a1y1, a2y1), 0.0f);
            float inter = iw * ih;
            float un = w1 * h1 + w2 * h2 - inter + EPSF;
            float iou = inter / un;
            float cw = fmaxf(a1x2, a2x2) - fminf(a1x1, a2x1);
            float ch = fmaxf(a1y2, a2y2) - fminf(a1y1, a2y1);
            float ca = cw * ch + EPSF;
            float giou = iou - (ca - un) / ca;
            gv = 1.0f - giou;
        }
    }
    float s1 = block_reduce256(l1v, red);
    __syncthreads();
    float s2 = block_reduce256(gv, red);
    if (threadIdx.x == 0) { l1part[blockIdx.x] = s1; gpart[blockIdx.x] = s2; }
}

// ---------------------------------------------------------------------------
// Final deterministic combine.
// ---------------------------------------------------------------------------
__global__ void k_final(const float* __restrict__ l1part, const float* __restrict__ gpart,
                        const float* __restrict__ lclspart,
                        const int* __restrict__ conf_any, const int* __restrict__ tn_ptr,
                        float* __restrict__ out, int nbox, int ncls) {
    if (threadIdx.x == 0 && blockIdx.x == 0) {
        float s1 = 0.0f, s2 = 0.0f, s3 = 0.0f;
        for (int i = 0; i < nbox; ++i) { s1 += l1part[i]; s2 += gpart[i]; }
        for (int i = 0; i < ncls; ++i) { s3 += lclspart[i]; }
        float tn = (float)(*tn_ptr);
        float loss = (5.0f * s1 + 2.0f * s2 + s3) / tn;
        out[0] = (*conf_any) ? loss : 0.0f;
    }
}

extern "C" void kernel_launch(void* const* d_in, const int* in_sizes, int n_in,
                              void* d_out, int out_size, void* d_ws, size_t ws_size,
                              hipStream_t stream) {
    const float* slog = (const float*)d_in[0];   // s_pred_logits (16,900,80)
    const float* sbox = (const float*)d_in[1];   // s_pred_boxes  (16,900,4)
    const float* tlog = (const float*)d_in[2];   // t_pred_logits (16,900,80)
    const float* tbox = (const float*)d_in[3];   // t_pred_boxes  (16,900,4)
    const int*   tn   = (const int*)d_in[4];     // target_nums scalar
    float* out = (float*)d_out;

    const int NBOX_BLK = (NROW + 255) / 256;     // 57
    const int NCLS_BLK = (NTILE + 7) / 8;        // 113

    char* w = (char*)d_ws;
    int*   max_idx  = (int*)w;   w += ((size_t)NROW * 4 + 63) & ~63ull;
    int*   iou_ok   = (int*)w;   w += ((size_t)NROW * 4 + 63) & ~63ull;
    int*   conf0    = (int*)w;   w += 64;
    int*   conf_any = (int*)w;   w += 64;
    float* l1part   = (float*)w; w += ((size_t)NBOX_BLK * 4 + 63) & ~63ull;
    float* gpart    = (float*)w; w += ((size_t)NBOX_BLK * 4 + 63) & ~63ull;
    float* lclspart = (float*)w; w += ((size_t)NCLS_BLK * 4 + 63) & ~63ull;
    float* cand_i   = (float*)w; w += ((size_t)KCH * NROW * 4 + 63) & ~63ull;
    float* cand_u   = (float*)w; w += ((size_t)KCH * NROW * 4 + 63) & ~63ull;
    int*   cand_j   = (int*)w;   w += ((size_t)KCH * NROW * 4 + 63) & ~63ull;

    k_conf0      <<<1, 32, 0, stream>>>(tlog, conf0, conf_any);
    k_confany    <<<((size_t)NROW * CC / 4 + 255) / 256, 256, 0, stream>>>(
                     (const float4*)tlog, conf_any);
    k_argmax_part<<<BB * 4 * KCH, 256, 0, stream>>>(sbox, tbox, cand_i, cand_u, cand_j);
    k_argmax_merge<<<NBOX_BLK, 256, 0, stream>>>(cand_i, cand_u, cand_j, max_idx, iou_ok);
    k_lcls       <<<NCLS_BLK, 256, 0, stream>>>(slog, tlog, max_idx, lclspart);
    k_box        <<<NBOX_BLK, 256, 0, stream>>>(sbox, tbox, max_idx, iou_ok, conf0,
                                                l1part, gpart);
    k_final      <<<1, 1, 0, stream>>>(l1part, gpart, lclspart, conf_any, tn, out,
                                       NBOX_BLK, NCLS_BLK);
}